// GAT_60902636257628
// MI455X (gfx1250) — compile-verified
//
#include <hip/hip_runtime.h>
#include <math.h>

// ---------------------------------------------------------------------------
// GAT (2-layer, heads=4, concat=False) for MI455X / gfx1250.
// GEMMs via v_wmma_f32_16x16x4_f32 (exact f32 path); edge softmax + scatter
// designed to stay resident in the 192MB L2.
// ---------------------------------------------------------------------------

typedef __attribute__((ext_vector_type(2))) float v2f;
typedef __attribute__((ext_vector_type(8))) float v8f;

#define HEADS 4
#define NEG_SLOPE 0.2f

// ---- f32 WMMA GEMM: C[M,Ncols] = A[M,K] * B[K,Ncols] -----------------------
// One wave32 per 16x16 output tile. M % 16 == 0, Ncols % 16 == 0, K % 4 == 0.
__global__ void gat_gemm_wmma_f32(const float* __restrict__ A,
                                  const float* __restrict__ B,
                                  float* __restrict__ C,
                                  int M, int K, int Ncols) {
    int wave   = (int)((blockIdx.x * blockDim.x + threadIdx.x) >> 5);
    int lane   = (int)(threadIdx.x & 31);
    int tilesN = Ncols >> 4;
    int tiles  = (M >> 4) * tilesN;
    if (wave >= tiles) return;                 // wave-uniform: EXEC stays all-1s

    int tm = wave / tilesN;
    int tn = wave - tm * tilesN;
    int m0 = tm << 4;
    int n0 = tn << 4;

    int r     = lane & 15;   // M-row (A) / N-col (B,C) within tile
    int khalf = lane >> 4;   // 0: K pair {0,1} ; 1: K pair {2,3}

    v8f acc = {};
    for (int k0 = 0; k0 < K; k0 += 4) {
        // A fragment: A[m0+r][k0 + 2*khalf + {0,1}]  (contiguous pair)
        const float* ap = A + (size_t)(m0 + r) * K + (k0 + 2 * khalf);
        v2f a;
        a.x = ap[0];
        a.y = ap[1];
        // B fragment: VGPR j = B[k0 + 2*khalf + j][n0 + r]
        const float* bp = B + (size_t)(k0 + 2 * khalf) * Ncols + (n0 + r);
        v2f b;
        b.x = bp[0];
        b.y = bp[Ncols];
        acc = __builtin_amdgcn_wmma_f32_16x16x4_f32(
            /*neg_a=*/false, a, /*neg_b=*/false, b,
            /*c_mod=*/(short)0, acc, /*reuse_a=*/false, /*reuse_b=*/false);
    }
    // C/D layout: VGPR j -> row (j + 8*khalf), col r
#pragma unroll
    for (int j = 0; j < 8; ++j) {
        C[(size_t)(m0 + j + 8 * khalf) * Ncols + (n0 + r)] = acc[j];
    }
}

// ---- per-(node,head) attention half-dots ------------------------------------
// xh layout: [N][HEADS][C].  i = n*HEADS + h.
__global__ void gat_attn_dots(const float* __restrict__ xh,
                              const float* __restrict__ a_src,
                              const float* __restrict__ a_dst,
                              float* __restrict__ als,
                              float* __restrict__ ald,
                              int NH, int C) {
    int i = (int)(blockIdx.x * blockDim.x + threadIdx.x);
    if (i >= NH) return;
    int h = i & (HEADS - 1);
    const float* xp = xh + (size_t)i * C;
    const float* as = a_src + (size_t)h * C;
    const float* ad = a_dst + (size_t)h * C;
    float s = 0.f, d = 0.f;
    for (int c = 0; c < C; ++c) {
        float v = xp[c];
        s += v * as[c];
        d += v * ad[c];
    }
    als[i] = s;
    ald[i] = d;
}

// ---- scratch re-init (graph-replay safe) ------------------------------------
__global__ void gat_init(float* __restrict__ agg, float* __restrict__ amax,
                         float* __restrict__ denom, int n256, int n4) {
    int i = (int)(blockIdx.x * blockDim.x + threadIdx.x);
    if (i < n256) agg[i] = 0.f;
    if (i < n4) {
        amax[i]  = -INFINITY;
        denom[i] = 0.f;
    }
}

// float atomic-max via signed-max / unsigned-min (correct across mixed signs)
__device__ __forceinline__ void atomicMaxFloat(float* addr, float v) {
    int iv = __float_as_int(v);
    if (iv >= 0)
        atomicMax((int*)addr, iv);
    else
        atomicMin((unsigned int*)addr, (unsigned int)iv);
}

__device__ __forceinline__ float leaky(float a) {
    return a >= 0.f ? a : NEG_SLOPE * a;
}

// ---- edge pass 1: segment max over destinations -----------------------------
// one thread per (edge, head); edges in [0,E) real, [E,E+N) self loops.
__global__ void gat_edge_max(const int* __restrict__ ei, int E, int N,
                             const float* __restrict__ als,
                             const float* __restrict__ ald,
                             float* __restrict__ amax) {
    int i = (int)(blockIdx.x * blockDim.x + threadIdx.x);
    int total = (E + N) * HEADS;
    if (i >= total) return;
    int e = i >> 2, h = i & 3;
    int src = (e < E) ? ei[e] : (e - E);
    int dst = (e < E) ? ei[E + e] : (e - E);
    float a = leaky(als[src * HEADS + h] + ald[dst * HEADS + h]);
    atomicMaxFloat(&amax[dst * HEADS + h], a);
}

// ---- edge pass 2: exp-sum denominators --------------------------------------
__global__ void gat_edge_expsum(const int* __restrict__ ei, int E, int N,
                                const float* __restrict__ als,
                                const float* __restrict__ ald,
                                const float* __restrict__ amax,
                                float* __restrict__ denom) {
    int i = (int)(blockIdx.x * blockDim.x + threadIdx.x);
    int total = (E + N) * HEADS;
    if (i >= total) return;
    int e = i >> 2, h = i & 3;
    int src = (e < E) ? ei[e] : (e - E);
    int dst = (e < E) ? ei[E + e] : (e - E);
    float a  = leaky(als[src * HEADS + h] + ald[dst * HEADS + h]);
    float ex = __expf(a - amax[dst * HEADS + h]);
    atomicAdd(&denom[dst * HEADS + h], ex);
}

// ---- edge pass 3: weighted scatter-aggregate --------------------------------
// one wave32 per edge; 256 channels = 8 per lane. L2-resident gather+atomics.
__global__ void gat_scatter(const int* __restrict__ ei, int E, int N,
                            const float* __restrict__ xh,
                            const float* __restrict__ als,
                            const float* __restrict__ ald,
                            const float* __restrict__ amax,
                            const float* __restrict__ denom,
                            float* __restrict__ agg, int C) {
    int e    = (int)((blockIdx.x * blockDim.x + threadIdx.x) >> 5);
    int lane = (int)(threadIdx.x & 31);
    if (e >= E + N) return;                    // wave-uniform
    int src = (e < E) ? ei[e] : (e - E);
    int dst = (e < E) ? ei[E + e] : (e - E);

    // lanes 0..3 each compute one head's attention weight, broadcast via shfl
    float aw = 0.f;
    if (lane < HEADS) {
        int h    = lane;
        float a  = leaky(als[src * HEADS + h] + ald[dst * HEADS + h]);
        float ex = __expf(a - amax[dst * HEADS + h]);
        aw       = ex / fmaxf(denom[dst * HEADS + h], 1e-16f);
    }
    float attn[HEADS];
#pragma unroll
    for (int h = 0; h < HEADS; ++h) attn[h] = __shfl(aw, h);

    const float* xs = xh + (size_t)src * (HEADS * C);
    float* od       = agg + (size_t)dst * (HEADS * C);
    // c = lane + 32*j  ->  head index = (c>>6) = (j>>1)  (C==64), constant per j
#pragma unroll
    for (int j = 0; j < 8; ++j) {
        int c = lane + 32 * j;
        atomicAdd(&od[c], xs[c] * attn[j >> 1]);
    }
}

// ---- head-mean + bias (+optional ELU) ---------------------------------------
__global__ void gat_finalize(const float* __restrict__ agg,
                             const float* __restrict__ bias,
                             float* __restrict__ out, int N, int C, int elu) {
    int i = (int)(blockIdx.x * blockDim.x + threadIdx.x);
    if (i >= N * C) return;
    int n = i / C, c = i - n * C;
    const float* ap = agg + (size_t)n * (HEADS * C) + c;
    float v = 0.25f * (ap[0] + ap[C] + ap[2 * C] + ap[3 * C]) + bias[c];
    if (elu) v = (v > 0.f) ? v : (expf(v) - 1.0f);
    out[i] = v;
}

// ---------------------------------------------------------------------------
extern "C" void kernel_launch(void* const* d_in, const int* in_sizes, int n_in,
                              void* d_out, int out_size, void* d_ws,
                              size_t ws_size, hipStream_t stream) {
    const float* x      = (const float*)d_in[0];
    const int*   ei     = (const int*)d_in[1];
    const float* W1     = (const float*)d_in[2];
    const float* a_src1 = (const float*)d_in[3];
    const float* a_dst1 = (const float*)d_in[4];
    const float* b1     = (const float*)d_in[5];
    const float* W2     = (const float*)d_in[6];
    const float* a_src2 = (const float*)d_in[7];
    const float* a_dst2 = (const float*)d_in[8];
    const float* b2     = (const float*)d_in[9];
    float* out          = (float*)d_out;

    const int IN = 128, C = 64;                 // H == OUT == 64
    const int N = in_sizes[0] / IN;             // 50000
    const int E = in_sizes[1] / 2;              // 800000
    const int NC  = HEADS * C;                  // 256
    const int ETOT = E + N;                     // edges incl. self loops

    // workspace layout (all f32)
    float* xh    = (float*)d_ws;                // N*256
    float* agg   = xh + (size_t)N * NC;         // N*256
    float* hbuf  = agg + (size_t)N * NC;        // N*64
    float* als   = hbuf + (size_t)N * C;        // N*4
    float* ald   = als + (size_t)N * HEADS;     // N*4
    float* amax  = ald + (size_t)N * HEADS;     // N*4
    float* denom = amax + (size_t)N * HEADS;    // N*4

    const int TB = 256;
    auto cdiv = [](long long a, long long b) { return (int)((a + b - 1) / b); };

    int gemm_blocks1 = cdiv((long long)(N / 16) * (NC / 16) * 32, TB);
    int dot_blocks   = cdiv((long long)N * HEADS, TB);
    int init_blocks  = cdiv((long long)N * NC, TB);
    int edge_blocks  = cdiv((long long)ETOT * HEADS, TB);
    int scat_blocks  = cdiv((long long)ETOT * 32, TB);
    int fin_blocks   = cdiv((long long)N * C, TB);

    // ---------------- layer 1: GATConv(128 -> 64, heads=4) ----------------
    gat_gemm_wmma_f32<<<gemm_blocks1, TB, 0, stream>>>(x, W1, xh, N, IN, NC);
    gat_attn_dots<<<dot_blocks, TB, 0, stream>>>(xh, a_src1, a_dst1, als, ald,
                                                 N * HEADS, C);
    gat_init<<<init_blocks, TB, 0, stream>>>(agg, amax, denom, N * NC, N * HEADS);
    gat_edge_max<<<edge_blocks, TB, 0, stream>>>(ei, E, N, als, ald, amax);
    gat_edge_expsum<<<edge_blocks, TB, 0, stream>>>(ei, E, N, als, ald, amax, denom);
    gat_scatter<<<scat_blocks, TB, 0, stream>>>(ei, E, N, xh, als, ald, amax,
                                                denom, agg, C);
    gat_finalize<<<fin_blocks, TB, 0, stream>>>(agg, b1, hbuf, N, C, /*elu=*/1);

    // ---------------- layer 2: GATConv(64 -> 64, heads=4) -----------------
    gat_gemm_wmma_f32<<<gemm_blocks1, TB, 0, stream>>>(hbuf, W2, xh, N, C, NC);
    gat_attn_dots<<<dot_blocks, TB, 0, stream>>>(xh, a_src2, a_dst2, als, ald,
                                                 N * HEADS, C);
    gat_init<<<init_blocks, TB, 0, stream>>>(agg, amax, denom, N * NC, N * HEADS);
    gat_edge_max<<<edge_blocks, TB, 0, stream>>>(ei, E, N, als, ald, amax);
    gat_edge_expsum<<<edge_blocks, TB, 0, stream>>>(ei, E, N, als, ald, amax, denom);
    gat_scatter<<<scat_blocks, TB, 0, stream>>>(ei, E, N, xh, als, ald, amax,
                                                denom, agg, C);
    gat_finalize<<<fin_blocks, TB, 0, stream>>>(agg, b2, out, N, C, /*elu=*/0);
}